// Compose_37091337568589
// MI455X (gfx1250) — compile-verified
//
#include <hip/hip_runtime.h>

// Compose displacement fields: phi(x) = right(x) + trilerp(left, x + right(x)),
// circulant (wrap) boundary. Shapes: [1, 3, 192, 192, 192] float32, channels-first.
//
// Memory-bound: ~255 MB min traffic -> ~11 us floor at 23.3 TB/s HBM. The 85 MB
// `left` field fits in MI455X's 192 MB L2, so the 24 gathers/voxel are
// L2-resident (and near-coalesced, since displacements are ~N(0,2)*2 voxels);
// `right`/`out` are streamed with non-temporal hints so they don't evict `left`.
//
// Launch geometry maps 1:1 onto the volume (block = 192 z-threads, grid =
// 192x192 y/x), eliminating all integer div/mod and the bounds guard.

namespace {

constexpr int DX = 192, DY = 192, DZ = 192;
constexpr int NVOX = DX * DY * DZ;  // 7,077,888 voxels per channel
constexpr int PFDIST = 8192;        // prefetch distance in floats (32 KB)

__device__ __forceinline__ int wrapi(int a, int n) {
  // floored modulo (matches jnp.mod for int32: result sign follows divisor)
  int m = a % n;
  return (m < 0) ? m + n : m;
}

__global__ __launch_bounds__(DZ) void compose_kernel(
    const float* __restrict__ left,
    const float* __restrict__ right,
    float* __restrict__ out) {
  const int z = threadIdx.x;   // 0..191, contiguous in memory
  const int y = blockIdx.x;    // 0..191
  const int x = blockIdx.y;    // 0..191
  const int idx = (x * DY + y) * DZ + z;

  // Speculative prefetch of the streaming 'right' channels ahead of this wave
  // (gfx1250 global_prefetch_b8). One lane per wave32 is enough: each address
  // pulls a whole cacheline; OOB speculative prefetches are silently dropped.
  if ((threadIdx.x & 31) == 0) {
    __builtin_prefetch(right + 0 * NVOX + idx + PFDIST, 0, 0);
    __builtin_prefetch(right + 1 * NVOX + idx + PFDIST, 0, 0);
    __builtin_prefetch(right + 2 * NVOX + idx + PFDIST, 0, 0);
  }

  // 'right' is read exactly once: non-temporal loads, don't pollute L2.
  const float rx = __builtin_nontemporal_load(right + 0 * NVOX + idx);
  const float ry = __builtin_nontemporal_load(right + 1 * NVOX + idx);
  const float rz = __builtin_nontemporal_load(right + 2 * NVOX + idx);

  const float cx = (float)x + rx;
  const float cy = (float)y + ry;
  const float cz = (float)z + rz;

  const float fx = floorf(cx), fy = floorf(cy), fz = floorf(cz);
  const float wx = cx - fx, wy = cy - fy, wz = cz - fz;
  const float ux = 1.0f - wx, uy = 1.0f - wy, uz = 1.0f - wz;

  const int x0 = wrapi((int)fx, DX);
  const int y0 = wrapi((int)fy, DY);
  const int z0 = wrapi((int)fz, DZ);
  const int x1 = (x0 + 1 == DX) ? 0 : x0 + 1;
  const int y1 = (y0 + 1 == DY) ? 0 : y0 + 1;
  const int z1 = (z0 + 1 == DZ) ? 0 : z0 + 1;

  // 8 corner flat indices (channel-local) in the reference accumulation order:
  // (dx,dy,dz) = 000,001,010,011,100,101,110,111
  const int b00 = (x0 * DY + y0) * DZ;
  const int b01 = (x0 * DY + y1) * DZ;
  const int b10 = (x1 * DY + y0) * DZ;
  const int b11 = (x1 * DY + y1) * DZ;

  int ci[8];
  ci[0] = b00 + z0; ci[1] = b00 + z1;
  ci[2] = b01 + z0; ci[3] = b01 + z1;
  ci[4] = b10 + z0; ci[5] = b10 + z1;
  ci[6] = b11 + z0; ci[7] = b11 + z1;

  float cw[8];
  cw[0] = ux * uy * uz; cw[1] = ux * uy * wz;
  cw[2] = ux * wy * uz; cw[3] = ux * wy * wz;
  cw[4] = wx * uy * uz; cw[5] = wx * uy * wz;
  cw[6] = wx * wy * uz; cw[7] = wx * wy * wz;

  const float r3[3] = {rx, ry, rz};
#pragma unroll
  for (int c = 0; c < 3; ++c) {
    // Gathers into 'left': regular (temporal) loads — the whole 85 MB field
    // stays resident in MI455X's 192 MB L2, so these are in-cache hits.
    const float* __restrict__ L = left + c * NVOX;
    float acc = 0.0f;
#pragma unroll
    for (int k = 0; k < 8; ++k) {
      acc = fmaf(cw[k], L[ci[k]], acc);
    }
    // Output is streamed once: non-temporal store.
    __builtin_nontemporal_store(r3[c] + acc, out + c * NVOX + idx);
  }
}

}  // namespace

extern "C" void kernel_launch(void* const* d_in, const int* in_sizes, int n_in,
                              void* d_out, int out_size, void* d_ws, size_t ws_size,
                              hipStream_t stream) {
  const float* left  = (const float*)d_in[0];
  const float* right = (const float*)d_in[1];
  float* out = (float*)d_out;

  dim3 block(DZ);        // 192 threads = 6 wave32s, fully coalesced z-row
  dim3 grid(DY, DX);     // 192 x 192 blocks, one per (x,y) row
  compose_kernel<<<grid, block, 0, stream>>>(left, right, out);
}